// ReadwiseHyena_64037962383574
// MI455X (gfx1250) — compile-verified
//
#include <hip/hip_runtime.h>
#include <cstdint>
#include <cstddef>

#define B_SZ 4
#define SEQ  4096
#define EMB  768
#define NORD 3
#define KS   3
#define NH   12
#define HD   64
#define CTOT 3072   // (NORD+1)*EMB
#define FILT_HID 64
#define FROWS 2304  // NORD*EMB

typedef float v2f __attribute__((ext_vector_type(2)));
typedef float v8f __attribute__((ext_vector_type(8)));

// ---------------------------------------------------------------------------
// CDNA5 async global->LDS copies (ASYNCcnt-tracked DMA path, no VGPR staging)
// ---------------------------------------------------------------------------
__device__ __forceinline__ void async_copy_b128(void* ldst, const void* gsrc) {
    asm volatile("global_load_async_to_lds_b128 %0, %1, off"
                 :: "v"((uint32_t)(uintptr_t)ldst), "v"(gsrc) : "memory");
}
__device__ __forceinline__ void async_copy_b32(void* ldst, const void* gsrc) {
    asm volatile("global_load_async_to_lds_b32 %0, %1, off"
                 :: "v"((uint32_t)(uintptr_t)ldst), "v"(gsrc) : "memory");
}
__device__ __forceinline__ void wait_async0() {
    asm volatile("s_wait_asynccnt 0x0" ::: "memory");
}

// ---------------------------------------------------------------------------
// Implicit filter MLP: filt[(ord*768+ch)][tl] for ord in {0,1}, tl < SEQ.
// z = sin(t @ W1 + b1); hmat = (z @ W2 + b2).T ; only first SEQ taps are used.
// ---------------------------------------------------------------------------
__global__ __launch_bounds__(256)
void hyena_filt_kernel(const float* __restrict__ t,
                       const float* __restrict__ W1, const float* __restrict__ b1,
                       const float* __restrict__ W2, const float* __restrict__ b2,
                       float* __restrict__ filt) {
    __shared__ float tsh[EMB];
    __shared__ float zsh[FILT_HID];
    int tl  = blockIdx.x;            // 0..SEQ-1
    int tid = threadIdx.x;
    for (int i = tid; i < EMB; i += 256) tsh[i] = t[(size_t)tl * EMB + i];
    __syncthreads();
    if (tid < FILT_HID) {
        float a = b1[tid];
        for (int e = 0; e < EMB; ++e) a += tsh[e] * W1[(size_t)e * FILT_HID + tid];
        zsh[tid] = sinf(a);
    }
    __syncthreads();
    // only ord 0 and 1 are consumed downstream -> 1536 rows
    for (int row = tid; row < 2 * EMB; row += 256) {
        float a = b2[row];
        #pragma unroll
        for (int f = 0; f < FILT_HID; ++f) a += zsh[f] * W2[(size_t)f * FROWS + row];
        filt[(size_t)row * SEQ + tl] = a;
    }
}

// ---------------------------------------------------------------------------
// fp32 WMMA GEMM: C = act(A(MxK) * B(KxN) + bias), row-major everywhere.
// 128 threads = 4 waves; block tile 64x64; K staged 16-deep in LDS via
// double-buffered GLOBAL_LOAD_ASYNC_TO_LDS_B128. Wave w computes rows
// [w*16, w*16+16) x 64 cols = 4 WMMA accumulators. act: 0 none, 1 sigmoid.
// LDS strides chosen conflict-free: A stride 20 words, B stride 80 words
// ((2*s-16) mod 64 in [16,48] for the half-wave split reads).
// ---------------------------------------------------------------------------
#define GM 64
#define GN 64
#define GK 16
#define APAD 4    // A row stride = GK+APAD = 20 words
#define BPAD 16   // B row stride = GN+BPAD = 80 words

__global__ __launch_bounds__(128)
void wmma_gemm_f32_kernel(const float* __restrict__ A, const float* __restrict__ Bm,
                          const float* __restrict__ bias, float* __restrict__ C,
                          int M, int N, int K, int act) {
    __shared__ float As[2][GM][GK + APAD];   // row-major: As[m][k]
    __shared__ float Bs[2][GK][GN + BPAD];   // Bs[k][n]
    int tid  = threadIdx.x;
    int wave = tid >> 5;
    int lane = tid & 31;
    int half = lane >> 4;        // 0: K+0/K+1, 1: K+2/K+3 (ISA f32 A/B layout)
    int lmod = lane & 15;
    int bm = blockIdx.y * GM;
    int bn = blockIdx.x * GN;

    v8f acc[4] = {};

    int a_row = tid >> 1;               // 0..63
    int a_kh  = (tid & 1) * 8;          // 0 or 8
    int b_kr  = tid >> 3;               // 0..15
    int b_nh  = (tid & 7) * 8;          // 0..56

    // issue one K-slice's async copies into LDS buffer `buf`
    auto issue = [&](int k0, int buf) {
        const float* asrc = A + (size_t)(bm + a_row) * K + k0 + a_kh;
        async_copy_b128(&As[buf][a_row][a_kh],     asrc);
        async_copy_b128(&As[buf][a_row][a_kh + 4], asrc + 4);
        const float* bsrc = Bm + (size_t)(k0 + b_kr) * N + bn + b_nh;
        async_copy_b128(&Bs[buf][b_kr][b_nh],     bsrc);
        async_copy_b128(&Bs[buf][b_kr][b_nh + 4], bsrc + 4);
    };

    issue(0, 0);
    int nk = K / GK;
    int mrow = wave * 16 + lmod;
    for (int ki = 0; ki < nk; ++ki) {
        int cur = ki & 1;
        wait_async0();       // copies into `cur` complete (this wave)
        __syncthreads();     // all waves' copies visible; prev compute done
        if (ki + 1 < nk) issue((ki + 1) * GK, cur ^ 1);   // overlap next slice
        #pragma unroll
        for (int kk = 0; kk < GK; kk += 4) {
            v2f a;           // contiguous pair -> single ds_load_b64
            a.x = As[cur][mrow][kk + 2 * half + 0];
            a.y = As[cur][mrow][kk + 2 * half + 1];
            #pragma unroll
            for (int nt = 0; nt < 4; ++nt) {
                v2f b;
                b.x = Bs[cur][kk + 2 * half + 0][nt * 16 + lmod];
                b.y = Bs[cur][kk + 2 * half + 1][nt * 16 + lmod];
                acc[nt] = __builtin_amdgcn_wmma_f32_16x16x4_f32(
                    false, a, false, b, (short)0, acc[nt], false, false);
            }
        }
    }

    int rbase = bm + wave * 16 + half * 8;   // C/D layout: VGPR j -> rows j / j+8
    #pragma unroll
    for (int nt = 0; nt < 4; ++nt) {
        int col = bn + nt * 16 + lmod;
        float bv = bias[col];
        #pragma unroll
        for (int j = 0; j < 8; ++j) {
            float v = acc[nt][j] + bv;
            if (act == 1) v = 1.0f / (1.0f + expf(-v));
            C[(size_t)(rbase + j) * N + col] = v;
        }
    }
}

// ---------------------------------------------------------------------------
// Depthwise conv (KS=3, zero pad 1) over S, then RoPE; u is (B*S, CTOT)
// row-major. Output SB[ord][b][ch][s], ch = nh*HD + hd, s contiguous.
// Each thread handles one (b, s, ord, hd-pair).
// ---------------------------------------------------------------------------
__global__ __launch_bounds__(256)
void conv_rope_kernel(const float* __restrict__ u, const float* __restrict__ cw,
                      const int* __restrict__ positions, float* __restrict__ SB) {
    size_t idx = (size_t)blockIdx.x * 256 + threadIdx.x;
    int jj = (int)(idx % 384); idx /= 384;      // hd-pair index within stream
    int ord = (int)(idx % 4);  idx /= 4;
    int s   = (int)(idx % SEQ); idx /= SEQ;
    int b   = (int)idx;
    int ch0 = jj * 2;                  // within stream [0,768)
    int c   = ord * EMB + ch0;         // within CTOT

    const float* up = u + ((size_t)(b * SEQ + s)) * CTOT + c;
    float w00 = cw[c * 3 + 0], w01 = cw[c * 3 + 1], w02 = cw[c * 3 + 2];
    float w10 = cw[(c + 1) * 3 + 0], w11 = cw[(c + 1) * 3 + 1], w12 = cw[(c + 1) * 3 + 2];

    float2 um = (s > 0)       ? *(const float2*)(up - CTOT) : make_float2(0.f, 0.f);
    float2 uc = *(const float2*)up;
    float2 un = (s < SEQ - 1) ? *(const float2*)(up + CTOT) : make_float2(0.f, 0.f);

    float y0 = w00 * um.x + w01 * uc.x + w02 * un.x;
    float y1 = w10 * um.y + w11 * uc.y + w12 * un.y;

    int jpos = jj & 31;                                 // freq index within head dim
    float inv = powf(10000.0f, -(float)(2 * jpos) / 64.0f);
    float ang = (float)positions[b * SEQ + s] * inv;
    float sn, cs;
    sincosf(ang, &sn, &cs);
    float o0 = y0 * cs - y1 * sn;
    float o1 = y0 * sn + y1 * cs;

    size_t obase = (((size_t)ord * B_SZ + b) * EMB + ch0) * SEQ + s;
    SB[obase]       = o0;
    SB[obase + SEQ] = o1;
}

// ---------------------------------------------------------------------------
// Elementwise product
// ---------------------------------------------------------------------------
__global__ __launch_bounds__(256)
void mult_kernel(const float* __restrict__ a, const float* __restrict__ b,
                 float* __restrict__ o, size_t n) {
    size_t i = (size_t)blockIdx.x * 256 + threadIdx.x;
    if (i < n) o[i] = a[i] * b[i];
}

// ---------------------------------------------------------------------------
// Causal long conv: y[r][s] = sum_{k<=s} h[ch][k] * v[r][s-k] + Bp[hd]*v[r][s]
// rows r = b*768+ch. Block: 1024 outputs (4/thread), LDS staged via async
// global->LDS b32 copies; negative v indices zero-filled with a plain LDS
// store under divergent EXEC (async loads honor EXEC per-lane).
// ---------------------------------------------------------------------------
#define CTS 1024
__global__ __launch_bounds__(256)
void causal_conv_kernel(const float* __restrict__ v, const float* __restrict__ filt,
                        const float* __restrict__ B_param, int ordi,
                        float* __restrict__ y) {
    __shared__ float hl[256];
    __shared__ float vl[CTS + 256];
    int r  = blockIdx.y;                  // 0..B*768-1
    int ch = r % EMB;
    int hd = ch % HD;
    int s0 = blockIdx.x * CTS;
    int t  = threadIdx.x;
    const float* vrow = v + (size_t)r * SEQ;
    const float* hrow = filt + ((size_t)ordi * EMB + ch) * SEQ;

    float acc[4] = {0.f, 0.f, 0.f, 0.f};

    for (int kt = 0; kt <= s0 + CTS - 1; kt += 256) {
        async_copy_b32(&hl[t], &hrow[kt + t]);
        int vbase = s0 - kt - 255;
        #pragma unroll
        for (int i = 0; i < 5; ++i) {
            int li = t + i * 256;
            int vi = vbase + li;
            if (vi >= 0) async_copy_b32(&vl[li], &vrow[vi]);
            else         vl[li] = 0.f;
        }
        wait_async0();
        __syncthreads();
        #pragma unroll 4
        for (int kk = 0; kk < 256; ++kk) {
            float hv = hl[kk];                  // uniform broadcast read
            int base = t + 255 - kk;
            acc[0] += hv * vl[base];
            acc[1] += hv * vl[base + 256];
            acc[2] += hv * vl[base + 512];
            acc[3] += hv * vl[base + 768];
        }
        __syncthreads();
    }
    float bp = B_param[ordi * HD + hd];
    #pragma unroll
    for (int i = 0; i < 4; ++i) {
        int si = s0 + t + i * 256;
        y[(size_t)r * SEQ + si] = acc[i] + bp * vrow[si];
    }
}

// ---------------------------------------------------------------------------
// v3 = y2 * x0; gelu(exact); g = gate * gelu  (transpose ch-major -> e-major)
// ---------------------------------------------------------------------------
__global__ __launch_bounds__(256)
void gelu_gate_kernel(const float* __restrict__ ybuf, const float* __restrict__ x0,
                      const float* __restrict__ gate, float* __restrict__ g) {
    size_t idx = (size_t)blockIdx.x * 256 + threadIdx.x;   // over B*768*SEQ
    int s = (int)(idx % SEQ); size_t rr = idx / SEQ;
    int ch = (int)(rr % EMB);
    int b  = (int)(rr / EMB);
    float v = ybuf[idx] * x0[idx];
    float ge = 0.5f * v * (1.0f + erff(v * 0.70710678118654752440f));
    size_t gi = ((size_t)(b * SEQ + s)) * EMB + ch;
    g[gi] = gate[gi] * ge;
}

// ---------------------------------------------------------------------------
// Launcher
// ---------------------------------------------------------------------------
extern "C" void kernel_launch(void* const* d_in, const int* in_sizes, int n_in,
                              void* d_out, int out_size, void* d_ws, size_t ws_size,
                              hipStream_t stream) {
    const float* emb      = (const float*)d_in[0];
    const int*   positions= (const int*)  d_in[1];
    const float* proj_w   = (const float*)d_in[2];
    const float* proj_b   = (const float*)d_in[3];
    const float* conv_w   = (const float*)d_in[4];
    const float* filt_w1  = (const float*)d_in[5];
    const float* filt_b1  = (const float*)d_in[6];
    const float* filt_w2  = (const float*)d_in[7];
    const float* filt_b2  = (const float*)d_in[8];
    const float* t        = (const float*)d_in[9];
    const float* B_param  = (const float*)d_in[10];
    const float* gate_w   = (const float*)d_in[11];
    const float* gate_b   = (const float*)d_in[12];
    const float* out_w    = (const float*)d_in[13];
    const float* out_b    = (const float*)d_in[14];
    float* out = (float*)d_out;
    float* ws  = (float*)d_ws;

    const size_t M    = (size_t)B_SZ * SEQ;          // 16384
    const size_t nU   = M * CTOT;                    // 50.3M
    const size_t nSB1 = (size_t)B_SZ * EMB * SEQ;    // 12.58M  (one stream)
    const size_t nSB  = 4 * nSB1;                    // 50.3M
    const size_t nF   = (size_t)2 * EMB * SEQ;       // 6.29M

    float* u    = ws;
    float* SB   = u + nU;
    float* filt = SB + nSB;
    float* vtmp = filt + nF;
    float* ybuf = vtmp + nSB1;
    float* gatebuf = u;                // u is dead after conv_rope
    float* gbuf    = u + nSB1;

    float* x0 = SB;                    // ord 0
    float* x1 = SB + 1 * nSB1;
    float* x2 = SB + 2 * nSB1;
    float* vstream = SB + 3 * nSB1;

    // 1) implicit filters (only ord 0/1, taps < SEQ)
    hyena_filt_kernel<<<SEQ, 256, 0, stream>>>(t, filt_w1, filt_b1, filt_w2, filt_b2, filt);

    // 2) u = emb @ proj_w + proj_b   (16384 x 3072 x 768)
    wmma_gemm_f32_kernel<<<dim3(CTOT / GN, M / GM), 128, 0, stream>>>(
        emb, proj_w, proj_b, u, (int)M, CTOT, EMB, 0);

    // 3) depthwise conv + RoPE -> SB streams
    {
        size_t total = (size_t)B_SZ * SEQ * 4 * 384;
        conv_rope_kernel<<<(unsigned)(total / 256), 256, 0, stream>>>(u, conv_w, positions, SB);
    }

    // 4) order-2 filter: vtmp = v * x2 ; ybuf = conv0(vtmp) + Bp0*vtmp
    mult_kernel<<<(unsigned)((nSB1 + 255) / 256), 256, 0, stream>>>(vstream, x2, vtmp, nSB1);
    causal_conv_kernel<<<dim3(SEQ / CTS, B_SZ * EMB), 256, 0, stream>>>(vtmp, filt, B_param, 0, ybuf);

    // 5) order-1 filter: vtmp = ybuf * x1 ; ybuf = conv1(vtmp) + Bp1*vtmp
    mult_kernel<<<(unsigned)((nSB1 + 255) / 256), 256, 0, stream>>>(ybuf, x1, vtmp, nSB1);
    causal_conv_kernel<<<dim3(SEQ / CTS, B_SZ * EMB), 256, 0, stream>>>(vtmp, filt, B_param, 1, ybuf);

    // 6) gate = sigmoid(emb @ gate_w + gate_b)
    wmma_gemm_f32_kernel<<<dim3(EMB / GN, M / GM), 128, 0, stream>>>(
        emb, gate_w, gate_b, gatebuf, (int)M, EMB, EMB, 1);

    // 7) g = gate * gelu(ybuf * x0)   (with ch-major -> e-major transpose)
    gelu_gate_kernel<<<(unsigned)((nSB1 + 255) / 256), 256, 0, stream>>>(ybuf, x0, gatebuf, gbuf);

    // 8) out = g @ out_w + out_b
    wmma_gemm_f32_kernel<<<dim3(EMB / GN, M / GM), 128, 0, stream>>>(
        gbuf, out_w, out_b, out, (int)M, EMB, EMB, 0);
}